// MambaBlock_27161373179954
// MI455X (gfx1250) — compile-verified
//
#include <hip/hip_runtime.h>
#include <hip/hip_bf16.h>
#include <cstdint>

// ---------------------------------------------------------------------------
// Mamba block for MI455X (gfx1250, wave32, WMMA + async/TDM data movement).
// GEMMs use fp32 V_WMMA_F32_16X16X4_F32 (workload is bandwidth-bound at
// ~55 GFLOP vs ~200MB of HBM traffic -> fp32 keeps exact reference parity).
// Tile staging is double-buffered: GLOBAL_LOAD_ASYNC_TO_LDS_B128 for A and
// the Tensor Data Mover for B (TDM pad_enable reproduces the 34-float padded
// LDS stride in hardware), overlapped with the WMMA compute of the previous
// tile via the split ASYNCcnt/TENSORcnt counters.
// ---------------------------------------------------------------------------

#define D_MODEL 1024
#define D_INNER 2048
#define D_STATE 16
#define DT_RANK 64
#define B_SZ    2
#define L_SEQ   2048
#define M_ROWS  (B_SZ * L_SEQ)   // 4096

typedef float v8f __attribute__((ext_vector_type(8)));
typedef float v2f __attribute__((ext_vector_type(2)));
typedef int   v4i __attribute__((ext_vector_type(4)));
typedef uint32_t u32x4v __attribute__((ext_vector_type(4)));
typedef int i32x4v __attribute__((ext_vector_type(4)));
typedef int i32x8v __attribute__((ext_vector_type(8)));

#define AS1 __attribute__((address_space(1)))
#define AS3 __attribute__((address_space(3)))

#if defined(__has_builtin)
#  if __has_builtin(__builtin_amdgcn_wmma_f32_16x16x4_f32)
#    define HAVE_WMMA_F32 1
#  endif
#  if __has_builtin(__builtin_amdgcn_global_load_async_to_lds_b128)
#    define HAVE_ASYNC 1
#  endif
#  if __has_builtin(__builtin_amdgcn_tensor_load_to_lds)
#    define HAVE_TDM 1
#  endif
#  if __has_builtin(__builtin_amdgcn_s_wait_asynccnt)
#    define HAVE_WAIT_ASYNC 1
#  endif
#  if __has_builtin(__builtin_amdgcn_s_wait_tensorcnt)
#    define HAVE_WAIT_TENSOR 1
#  endif
#endif
#ifndef HAVE_WMMA_F32
#define HAVE_WMMA_F32 0
#endif
#ifndef HAVE_ASYNC
#define HAVE_ASYNC 0
#endif
#ifndef HAVE_TDM
#define HAVE_TDM 0
#endif
#ifndef HAVE_WAIT_ASYNC
#define HAVE_WAIT_ASYNC 0
#endif
#ifndef HAVE_WAIT_TENSOR
#define HAVE_WAIT_TENSOR 0
#endif

__device__ __forceinline__ void wait_async0() {
#if HAVE_WAIT_ASYNC
    __builtin_amdgcn_s_wait_asynccnt(0);
#else
    asm volatile("s_wait_asynccnt 0x0" ::: "memory");
#endif
}
__device__ __forceinline__ void wait_tensor0() {
#if HAVE_WAIT_TENSOR
    __builtin_amdgcn_s_wait_tensorcnt(0);
#else
    asm volatile("s_wait_tensorcnt 0x0" ::: "memory");
#endif
}

__device__ __forceinline__ float silu_f(float v) {
    return v / (1.0f + expf(-v));
}
__device__ __forceinline__ float softplus_f(float v) {
    return fmaxf(v, 0.0f) + log1pf(expf(-fabsf(v)));
}

#if HAVE_TDM
// ---------------------------------------------------------------------------
// Issue one TDM descriptor: load a 64-row x 32-col fp32 tile (rows = N dim,
// cols = K dim, row stride ld_elems) into LDS with a 2-DWORD pad after every
// 32 DWORDs -> LDS row stride 34 floats. Rows/cols beyond rem_n/rem_k
// zero-fill (ISA: OOB tensor reads return zero), replacing guard branches.
// D# layout per CDNA5 ISA §8.3/§8.4. Wave-level op; call from one wave only.
// ---------------------------------------------------------------------------
__device__ __forceinline__ void tdm_load_tile_f32(
    const float* gbase, uint32_t lds_byte_addr,
    uint32_t rem_k, uint32_t rem_n, uint32_t ld_elems)
{
    const uint64_t ga = (uint64_t)(uintptr_t)gbase;
    u32x4v g0;
    g0[0] = 1u;                                              // count=1, user D#
    g0[1] = lds_byte_addr;                                   // lds_addr
    g0[2] = (uint32_t)ga;                                    // global_addr[31:0]
    g0[3] = (uint32_t)((ga >> 32) & 0x01FFFFFFu) | (2u << 30); // [56:32] | type=2
    i32x8v g1;
    g1[0] = (int)((2u << 16)      // data_size = 4 bytes
                | (1u << 20)      // pad_enable
                | (4u << 22)      // pad_interval code 4 = 32 DWORDs
                | (1u << 25));    // pad_amount  code 1 = 2 DWORDs
    g1[1] = (int)((rem_k & 0xFFFFu) << 16);                  // tensor_dim0 lo
    g1[2] = (int)((rem_k >> 16) | ((rem_n & 0xFFFFu) << 16));// dim0 hi | dim1 lo
    g1[3] = (int)((rem_n >> 16) | (32u << 16));              // dim1 hi | tile_dim0=32
    g1[4] = (int)64u;                                        // tile_dim1=64, tile_dim2=0
    g1[5] = (int)ld_elems;                                   // tensor_dim0_stride lo
    g1[6] = 0;                                               // stride hi | dim1_stride lo
    g1[7] = 0;
    const i32x4v z4 = {0, 0, 0, 0};
#if defined(__clang_major__) && __clang_major__ >= 23
    const i32x8v z8 = {0, 0, 0, 0, 0, 0, 0, 0};
    __builtin_amdgcn_tensor_load_to_lds(g0, g1, z4, z4, z8, 0);
#else
    __builtin_amdgcn_tensor_load_to_lds(g0, g1, z4, z4, 0);
#endif
}
#endif // HAVE_TDM

// ---------------------------------------------------------------------------
// Generic NT GEMM: C[m,n] = sum_k A[m*lda+k] * B[n*ldb+k]   (both K-major)
// Requires M % 64 == 0 and K % 32 == 0 (all call sites satisfy this).
// Block tile 64x64, BK=32, LDS stride 34 (conflict-free float2 frag reads),
// DOUBLE-BUFFERED: tile i+1's async/TDM transfers are issued right after the
// barrier that retires tile i's transfers, then run concurrently with tile
// i's 16 WMMAs. 8 waves = 4(m) x 2(n); each wave owns a 16x32 C strip.
// EPI: 0 plain, 1 softplus(acc+bias[n]), 2 acc+resid[m,n].
// GN: guard B-tile rows against N (manual-load fallback only; the TDM path
// zero-fills OOB rows via tensor_dim1).
// ---------------------------------------------------------------------------
#define BK   32
#define LSTR 34

template <int EPI, bool GN>
__global__ __launch_bounds__(256) void gemm_nt_f32(
    const float* __restrict__ A, const float* __restrict__ B,
    float* __restrict__ C, int M, int N, int K, int lda, int ldb, int ldc,
    const float* __restrict__ bias, const float* __restrict__ resid, int res_ld)
{
    __shared__ float As[2][64 * LSTR];
    __shared__ float Bs[2][64 * LSTR];

    const int tid   = threadIdx.x;
    const int lane  = tid & 31;
    const int wave  = tid >> 5;
    const int wrow  = (wave >> 1) << 4;   // 0,16,32,48
    const int wcol  = (wave &  1) << 5;   // 0,32
    const int r15   = lane & 15;
    const int rhalf = lane >> 4;          // 0/1: K-pair / M-half selector

    const int m0 = blockIdx.y << 6;
    const int n0 = blockIdx.x << 6;
    const uint32_t rem_n = (uint32_t)(N > n0 ? N - n0 : 0);

    // issue all transfers for one K-tile into LDS buffer `buf`
    auto stage = [&](int k0, int buf) {
#if HAVE_TDM
        if (wave == 0)  // B tile via Tensor Data Mover (one wave per block)
            tdm_load_tile_f32(&B[(size_t)n0 * ldb + k0],
                              (uint32_t)(uintptr_t)(void*)&Bs[buf][0],
                              (uint32_t)(K - k0), rem_n, (uint32_t)ldb);
#endif
#pragma unroll
        for (int i = 0; i < 2; ++i) {
            const int lin = (i << 8) + tid;       // 0..511
            const int r   = lin >> 3;             // 0..63
            const int kq  = (lin & 7) << 2;       // 0..28 step 4
#if HAVE_ASYNC
            __builtin_amdgcn_global_load_async_to_lds_b128(
                (AS1 v4i*)&A[(size_t)(m0 + r) * lda + k0 + kq],
                (AS3 v4i*)&As[buf][r * LSTR + kq], 0, 0);
#else
            const float4 av = *(const float4*)&A[(size_t)(m0 + r) * lda + k0 + kq];
            *(float2*)&As[buf][r * LSTR + kq]     = make_float2(av.x, av.y);
            *(float2*)&As[buf][r * LSTR + kq + 2] = make_float2(av.z, av.w);
#endif
#if !HAVE_TDM
            float4 bv = make_float4(0.f, 0.f, 0.f, 0.f);
            if (!GN || (n0 + r < N))
                bv = *(const float4*)&B[(size_t)(n0 + r) * ldb + k0 + kq];
            *(float2*)&Bs[buf][r * LSTR + kq]     = make_float2(bv.x, bv.y);
            *(float2*)&Bs[buf][r * LSTR + kq + 2] = make_float2(bv.z, bv.w);
#endif
        }
    };

    v8f acc0 = {0, 0, 0, 0, 0, 0, 0, 0};
    v8f acc1 = {0, 0, 0, 0, 0, 0, 0, 0};

    const int ntiles = K / BK;
    stage(0, 0);                              // prologue: tile 0 in flight

    for (int ti = 0; ti < ntiles; ++ti) {
#if HAVE_ASYNC
        wait_async0();                        // tile ti's A transfers done
#endif
#if HAVE_TDM
        if (wave == 0) wait_tensor0();        // tile ti's B DMA done
#endif
        // Barrier: (a) tile ti's LDS contents visible to all waves;
        // (b) every wave finished computing tile ti-1, so the buffer we are
        //     about to overwrite with tile ti+1 is no longer being read.
        __syncthreads();

        if (ti + 1 < ntiles)                  // overlap: next tile's DMA now
            stage((ti + 1) * BK, (ti + 1) & 1);

        const float* __restrict__ Ab = &As[ti & 1][0];
        const float* __restrict__ Bb = &Bs[ti & 1][0];
        const int aBase  = (wrow + r15) * LSTR;
        const int b0Base = (wcol + r15) * LSTR;
        const int b1Base = (wcol + 16 + r15) * LSTR;

#pragma unroll
        for (int kk = 0; kk < BK; kk += 4) {
#if HAVE_WMMA_F32
            // ISA 7.12.2: A 16x4 -> lane half picks K-pair {0,1}|{2,3};
            // B 4x16 -> N = lane&15, VGPR v holds K-row v (+2 upper half).
            const int ko = kk + (rhalf << 1);
            v2f a, b0v, b1v;
            a.x   = Ab[aBase + ko];      a.y   = Ab[aBase + ko + 1];
            b0v.x = Bb[b0Base + ko];     b0v.y = Bb[b0Base + ko + 1];
            b1v.x = Bb[b1Base + ko];     b1v.y = Bb[b1Base + ko + 1];
            acc0 = __builtin_amdgcn_wmma_f32_16x16x4_f32(
                false, a, false, b0v, (short)0, acc0, false, false);
            acc1 = __builtin_amdgcn_wmma_f32_16x16x4_f32(
                false, a, false, b1v, (short)0, acc1, false, false);
#else
            // layout-identical scalar fallback (C frag: VGPR v -> row v+8*rhalf)
#pragma unroll
            for (int v = 0; v < 8; ++v) {
                const int rr = (wrow + v + (rhalf << 3)) * LSTR;
#pragma unroll
                for (int k = 0; k < 4; ++k) {
                    const float av = Ab[rr + kk + k];
                    acc0[v] += av * Bb[b0Base + kk + k];
                    acc1[v] += av * Bb[b1Base + kk + k];
                }
            }
#endif
        }
    }

    // epilogue: C/D frag layout: VGPR v -> M = wrow+v+8*rhalf, N = lane&15
#pragma unroll
    for (int v = 0; v < 8; ++v) {
        const int gr  = m0 + wrow + v + (rhalf << 3);
        const int gc0 = n0 + wcol + r15;
        const int gc1 = gc0 + 16;
        if (gc0 < N) {
            float r0 = acc0[v];
            if (EPI == 1)      r0 = softplus_f(r0 + bias[gc0]);
            else if (EPI == 2) r0 += resid[(size_t)gr * res_ld + gc0];
            C[(size_t)gr * ldc + gc0] = r0;
        }
        if (gc1 < N) {
            float r1 = acc1[v];
            if (EPI == 1)      r1 = softplus_f(r1 + bias[gc1]);
            else if (EPI == 2) r1 += resid[(size_t)gr * res_ld + gc1];
            C[(size_t)gr * ldc + gc1] = r1;
        }
    }
}

// ---------------------------------------------------------------------------
// Causal depthwise conv (width 4) + bias + SiLU over the u half of xz.
// ---------------------------------------------------------------------------
__global__ __launch_bounds__(256) void conv_silu_kernel(
    const float* __restrict__ xz, const float* __restrict__ cw,
    const float* __restrict__ cb, float* __restrict__ uc)
{
    const int idx = blockIdx.x * 256 + threadIdx.x;   // < 4096*2048
    const int d   = idx & (D_INNER - 1);
    const int m   = idx >> 11;
    const int t   = m & (L_SEQ - 1);
    const int bm  = m - t;                            // b * L_SEQ
    float acc = cb[d];
#pragma unroll
    for (int j = 0; j < 4; ++j) {
        const int tt = t + j - 3;
        if (tt >= 0)
            acc += cw[(d << 2) + j] * xz[((size_t)(bm + tt) << 12) + d];
    }
    uc[(size_t)m * D_INNER + d] = silu_f(acc);
}

// ---------------------------------------------------------------------------
// Selective scan. One thread per (b, d): 16 states in registers, sequential
// over t. Shared B/C rows double-buffered through LDS via async-to-LDS in
// 64-timestep chunks (broadcast reads -> conflict-free); the next chunk's
// transfers overlap the current chunk's recurrence. Epilogue fuses
// y = (scan + u*D) * silu(z), overwriting the dt buffer in place.
// ---------------------------------------------------------------------------
#define TCHUNK 64

__global__ __launch_bounds__(256) void mamba_scan_kernel(
    float* __restrict__ dtY,            // in: dt ; out: gated y (in-place)
    const float* __restrict__ uc,
    const float* __restrict__ xz,
    const float* __restrict__ xdbl,
    const float* __restrict__ A_log,
    const float* __restrict__ Dvec)
{
    __shared__ float BC[2][TCHUNK][32]; // [buf][t][0..15]=B, [16..31]=C

    const int tid = threadIdx.x;
    const int b   = blockIdx.x >> 3;
    const int d   = ((blockIdx.x & 7) << 8) + tid;
    const int mbase = b << 11;          // b * L_SEQ

    auto stage = [&](int t0, int buf) {
#pragma unroll
        for (int i = 0; i < 2; ++i) {
            const int lin = (i << 8) + tid;   // float4 index 0..511
            const int tl  = lin >> 3;
            const int c4  = (lin & 7) << 2;
#if HAVE_ASYNC
            __builtin_amdgcn_global_load_async_to_lds_b128(
                (AS1 v4i*)&xdbl[(size_t)(mbase + t0 + tl) * 96 + DT_RANK + c4],
                (AS3 v4i*)&BC[buf][tl][c4], 0, 0);
#else
            *(float4*)&BC[buf][tl][c4] =
                *(const float4*)&xdbl[(size_t)(mbase + t0 + tl) * 96 + DT_RANK + c4];
#endif
        }
    };

    float An[D_STATE];
#pragma unroll
    for (int n = 0; n < D_STATE; ++n)
        An[n] = -expf(A_log[(size_t)d * D_STATE + n]);
    const float Dd = Dvec[d];

    float h[D_STATE];
#pragma unroll
    for (int n = 0; n < D_STATE; ++n) h[n] = 0.0f;

    const int nchunks = L_SEQ / TCHUNK;
    stage(0, 0);

    for (int c = 0; c < nchunks; ++c) {
#if HAVE_ASYNC
        wait_async0();
#endif
        __syncthreads();   // chunk c visible; previous buffer free to overwrite

        if (c + 1 < nchunks)
            stage((c + 1) * TCHUNK, (c + 1) & 1);

        const int t0 = c * TCHUNK;
        for (int tl = 0; tl < TCHUNK; ++tl) {
            const size_t idx = (size_t)(mbase + t0 + tl) * D_INNER + d;
            const float dt_t = dtY[idx];
            const float u_t  = uc[idx];
            const float du   = dt_t * u_t;
            float y = 0.0f;
#pragma unroll
            for (int n = 0; n < D_STATE; ++n) {
                const float dA = expf(dt_t * An[n]);
                h[n] = h[n] * dA + du * BC[c & 1][tl][n];
                y += h[n] * BC[c & 1][tl][16 + n];
            }
            const float zv = xz[((size_t)(mbase + t0 + tl) << 12) + D_INNER + d];
            dtY[idx] = (y + u_t * Dd) * silu_f(zv);
        }
    }
}

// ---------------------------------------------------------------------------
// Rowwise LayerNorm (residual already folded in by the out_proj epilogue).
// ---------------------------------------------------------------------------
__global__ __launch_bounds__(256) void layernorm_kernel(
    const float* __restrict__ y, const float* __restrict__ w,
    const float* __restrict__ bb, float* __restrict__ out)
{
    __shared__ float s1[256];
    __shared__ float s2[256];
    const int row = blockIdx.x;
    const int tid = threadIdx.x;

    const float4 v = ((const float4*)(y + (size_t)row * D_MODEL))[tid];
    s1[tid] = v.x + v.y + v.z + v.w;
    s2[tid] = v.x * v.x + v.y * v.y + v.z * v.z + v.w * v.w;
    __syncthreads();
    for (int st = 128; st > 0; st >>= 1) {
        if (tid < st) { s1[tid] += s1[tid + st]; s2[tid] += s2[tid + st]; }
        __syncthreads();
    }
    const float mu  = s1[0] * (1.0f / D_MODEL);
    const float var = s2[0] * (1.0f / D_MODEL) - mu * mu;
    const float inv = rsqrtf(var + 1e-5f);

    const float4 wv = ((const float4*)w)[tid];
    const float4 bv = ((const float4*)bb)[tid];
    float4 o;
    o.x = (v.x - mu) * inv * wv.x + bv.x;
    o.y = (v.y - mu) * inv * wv.y + bv.y;
    o.z = (v.z - mu) * inv * wv.z + bv.z;
    o.w = (v.w - mu) * inv * wv.w + bv.w;
    ((float4*)(out + (size_t)row * D_MODEL))[tid] = o;
}

// ---------------------------------------------------------------------------
extern "C" void kernel_launch(void* const* d_in, const int* in_sizes, int n_in,
                              void* d_out, int out_size, void* d_ws, size_t ws_size,
                              hipStream_t stream)
{
    const float* x         = (const float*)d_in[0];
    const float* in_proj_w = (const float*)d_in[1];
    const float* conv_w    = (const float*)d_in[2];
    const float* conv_b    = (const float*)d_in[3];
    const float* x_proj_w  = (const float*)d_in[4];
    const float* dt_proj_w = (const float*)d_in[5];
    const float* dt_proj_b = (const float*)d_in[6];
    const float* A_log     = (const float*)d_in[7];
    const float* Dvec      = (const float*)d_in[8];
    const float* out_proj  = (const float*)d_in[9];
    const float* ln_w      = (const float*)d_in[10];
    const float* ln_b      = (const float*)d_in[11];
    float* out = (float*)d_out;

    // workspace layout (floats): every buffer written before read each call
    float* ws   = (float*)d_ws;
    float* xz   = ws;                                        // 4096 x 4096
    float* uc   = xz   + (size_t)M_ROWS * 2 * D_INNER;       // 4096 x 2048
    float* xdbl = uc   + (size_t)M_ROWS * D_INNER;           // 4096 x 96
    float* dtY  = xdbl + (size_t)M_ROWS * 96;                // 4096 x 2048 (dt -> y)
    float* ypre = dtY  + (size_t)M_ROWS * D_INNER;           // 4096 x 1024

    const dim3 blk(256);

    // 1) xz = x @ in_proj_w^T        (4096 x 4096, K=1024)
    gemm_nt_f32<0, false><<<dim3(64, 64), blk, 0, stream>>>(
        x, in_proj_w, xz, M_ROWS, 2 * D_INNER, D_MODEL,
        D_MODEL, D_MODEL, 2 * D_INNER, nullptr, nullptr, 0);

    // 2) u_c = silu(depthwise_conv(u) + conv_b)
    conv_silu_kernel<<<(M_ROWS * D_INNER) / 256, blk, 0, stream>>>(
        xz, conv_w, conv_b, uc);

    // 3) x_dbl = u_c @ x_proj_w^T    (4096 x 96, K=2048)
    gemm_nt_f32<0, true><<<dim3(2, 64), blk, 0, stream>>>(
        uc, x_proj_w, xdbl, M_ROWS, DT_RANK + 2 * D_STATE, D_INNER,
        D_INNER, D_INNER, 96, nullptr, nullptr, 0);

    // 4) dt = softplus(dt_lr @ dt_proj_w^T + dt_proj_b)  (4096 x 2048, K=64)
    gemm_nt_f32<1, false><<<dim3(32, 64), blk, 0, stream>>>(
        xdbl, dt_proj_w, dtY, M_ROWS, D_INNER, DT_RANK,
        96, DT_RANK, D_INNER, dt_proj_b, nullptr, 0);

    // 5) selective scan + gating, in-place over the dt buffer
    mamba_scan_kernel<<<B_SZ * (D_INNER / 256), blk, 0, stream>>>(
        dtY, uc, xz, xdbl, A_log, Dvec);

    // 6) y_pre = y @ out_proj^T + x  (4096 x 1024, K=2048, fused residual)
    gemm_nt_f32<2, false><<<dim3(16, 64), blk, 0, stream>>>(
        dtY, out_proj, ypre, M_ROWS, D_MODEL, D_INNER,
        D_INNER, D_INNER, D_MODEL, nullptr, x, D_MODEL);

    // 7) LayerNorm -> d_out
    layernorm_kernel<<<M_ROWS, blk, 0, stream>>>(ypre, ln_w, ln_b, out);
}